// stochastic_opt_model_2499670966486
// MI455X (gfx1250) — compile-verified
//
#include <hip/hip_runtime.h>
#include <math.h>

// Problem constants (fixed by the reference)
static constexpr int kN     = 24;   // decision vector length
static constexpr int kM     = 23;   // ramp constraints
static constexpr int kOuter = 10;
static constexpr int kIP    = 12;
static constexpr int kBlk   = 64;   // threads per block (2 waves)

#define AS_GLOBAL __attribute__((address_space(1)))
#define AS_LDS    __attribute__((address_space(3)))

typedef int v4i __attribute__((__vector_size__(16)));

#if __has_builtin(__builtin_amdgcn_global_load_async_to_lds_b128)
#define USE_ASYNC_LDS 1
#else
#define USE_ASYNC_LDS 0
#endif

// Hardware v_rcp_f32 (~1 ulp). Safe here: IPM recomputes exact residuals each
// step and H is strictly diagonally dominant, so the solve self-corrects.
__device__ __forceinline__ float rcp(float a) { return __builtin_amdgcn_rcpf(a); }

__global__ __launch_bounds__(kBlk)
void stoch_opt_ipm_kernel(const float* __restrict__ xin,
                          const float* __restrict__ muin,
                          const float* __restrict__ sgin,
                          float* __restrict__ out, int B)
{
    const int tid = threadIdx.x;
    const int i   = blockIdx.x * kBlk + tid;
    if (i >= B) return;

    const float GU          = 50.0f;
    const float coef        = 50.5f;            // gamma_over + gamma_under
    const float h           = 0.4f;             // c_ramp
    const float INV_SQRT2   = 0.7071067811865476f;
    const float INV_SQRT2PI = 0.3989422804014327f;

    const size_t base = (size_t)i * kN;

    // Per-lane register-resident state. All length-24/23 loops fully unroll so
    // every array index is a compile-time constant -> pure VGPR arrays.
    float x[kN], mu[kN], rsig[kN], q[kN], p[kN], s[kM], z[kM];

#if USE_ASYNC_LDS
    // CDNA5 async copy path: GLOBAL_LOAD_ASYNC_TO_LDS_B128 (ASYNCcnt-tracked),
    // then s_wait_asynccnt 0 and ds_load_b128. Each thread consumes only the
    // 96B its own wave issued, so no workgroup barrier is required.
    __shared__ __attribute__((aligned(16))) float lx[kBlk * kN];
    __shared__ __attribute__((aligned(16))) float lm[kBlk * kN];
    __shared__ __attribute__((aligned(16))) float ls[kBlk * kN];
    {
        AS_GLOBAL v4i* gx = (AS_GLOBAL v4i*)(xin  + base);
        AS_GLOBAL v4i* gm = (AS_GLOBAL v4i*)(muin + base);
        AS_GLOBAL v4i* gs = (AS_GLOBAL v4i*)(sgin + base);
        AS_LDS    v4i* dx_ = (AS_LDS v4i*)&lx[tid * kN];
        AS_LDS    v4i* dm_ = (AS_LDS v4i*)&lm[tid * kN];
        AS_LDS    v4i* ds_ = (AS_LDS v4i*)&ls[tid * kN];
        #pragma unroll
        for (int c = 0; c < 6; ++c) {
            __builtin_amdgcn_global_load_async_to_lds_b128(gx + c, dx_ + c, 0, 0);
            __builtin_amdgcn_global_load_async_to_lds_b128(gm + c, dm_ + c, 0, 0);
            __builtin_amdgcn_global_load_async_to_lds_b128(gs + c, ds_ + c, 0, 0);
        }
        asm volatile("s_wait_asynccnt 0x0" ::: "memory");
        const float4* xv = (const float4*)&lx[tid * kN];
        const float4* mv = (const float4*)&lm[tid * kN];
        const float4* sv = (const float4*)&ls[tid * kN];
        #pragma unroll
        for (int v = 0; v < 6; ++v) {
            float4 a = xv[v], b = mv[v], c = sv[v];
            x[4*v+0]=a.x; x[4*v+1]=a.y; x[4*v+2]=a.z; x[4*v+3]=a.w;
            mu[4*v+0]=b.x; mu[4*v+1]=b.y; mu[4*v+2]=b.z; mu[4*v+3]=b.w;
            rsig[4*v+0]=1.0f/c.x; rsig[4*v+1]=1.0f/c.y;
            rsig[4*v+2]=1.0f/c.z; rsig[4*v+3]=1.0f/c.w;
        }
    }
#else
    // Direct path: 96B per lane, 16B-aligned -> global_load_b128
    __builtin_prefetch(xin  + base, 0, 0);
    __builtin_prefetch(muin + base, 0, 0);
    __builtin_prefetch(sgin + base, 0, 0);
    {
        const float4* xv = (const float4*)(xin  + base);
        const float4* mv = (const float4*)(muin + base);
        const float4* sv = (const float4*)(sgin + base);
        #pragma unroll
        for (int v = 0; v < 6; ++v) {
            float4 a = xv[v], b = mv[v], c = sv[v];
            x[4*v+0]=a.x; x[4*v+1]=a.y; x[4*v+2]=a.z; x[4*v+3]=a.w;
            mu[4*v+0]=b.x; mu[4*v+1]=b.y; mu[4*v+2]=b.z; mu[4*v+3]=b.w;
            rsig[4*v+0]=1.0f/c.x; rsig[4*v+1]=1.0f/c.y;
            rsig[4*v+2]=1.0f/c.z; rsig[4*v+3]=1.0f/c.w;
        }
    }
#endif

    for (int outer = 0; outer < kOuter; ++outer) {
        // --- q, p from current x (Gaussian pdf/cdf) ---
        #pragma unroll
        for (int j = 0; j < kN; ++j) {
            float t   = (x[j] - mu[j]) * rsig[j];
            float pdf = expf(-0.5f * t * t) * INV_SQRT2PI;
            q[j] = pdf * rsig[j] * coef + 1.0f;
            float cdf = 0.5f * (1.0f + erff(t * INV_SQRT2));
            p[j] = cdf * coef - GU - mu[j];
        }
        // --- reset IPM state (x=0 strictly feasible, s=h, z=1) ---
        #pragma unroll
        for (int j = 0; j < kN; ++j) x[j] = 0.0f;
        #pragma unroll
        for (int k = 0; k < kM; ++k) { s[k] = h; z[k] = 1.0f; }

        for (int it = 0; it < kIP; ++it) {
            float rp[kM], rc[kM], rs[kM], e[kM];
            float d[kN], rhs[kN], cp[kN], dx[kN];

            // centering parameter: 0.1 * dot(s,z)/m
            float dot = 0.0f;
            #pragma unroll
            for (int k = 0; k < kM; ++k) dot += s[k] * z[k];
            const float sigmu = 0.1f * dot * (1.0f / (float)kM);

            #pragma unroll
            for (int k = 0; k < kM; ++k) {
                rp[k] = (x[k+1] - x[k]) + s[k] - h;   // Gx + s - h
                rc[k] = s[k] * z[k] - sigmu;          // s*z - 0.1*mu
                rs[k] = rcp(s[k]);                    // v_rcp_f32
            }

            // H = diag(q) + G^T diag(w) G is SYMMETRIC TRIDIAGONAL:
            //   d[j] = q_j + w_j[j<M] + w_{j-1}[j>=1],  e[j] = H[j,j+1] = -w_j
            #pragma unroll
            for (int j = 0; j < kN; ++j) {
                float wj  = (j < kM) ? z[j]   * rs[j]   : 0.0f;
                float wjm = (j >= 1) ? z[j-1] * rs[j-1] : 0.0f;
                d[j] = q[j] + wj + wjm;
                if (j < kM) e[j] = -wj;
            }

            // rhs = -r_d - G^T u,  r_d = q*x + p + G^T z,  u = (z*rp - rc)/s
            #pragma unroll
            for (int j = 0; j < kN; ++j) {
                float gtz = ((j >= 1) ? z[j-1] : 0.0f) - ((j < kM) ? z[j] : 0.0f);
                float rd  = q[j] * x[j] + p[j] + gtz;
                float uj  = (j < kM) ? (z[j]   * rp[j]   - rc[j])   * rs[j]   : 0.0f;
                float ujm = (j >= 1) ? (z[j-1] * rp[j-1] - rc[j-1]) * rs[j-1] : 0.0f;
                rhs[j] = -rd - (ujm - uj);
            }

            // Thomas solve (H strictly diagonally dominant: q>=1) -> dx
            {
                float inv = rcp(d[0]);
                cp[0]  = e[0] * inv;
                rhs[0] = rhs[0] * inv;
                #pragma unroll
                for (int j = 1; j < kN; ++j) {
                    float m2 = rcp(d[j] - e[j-1] * cp[j-1]);
                    cp[j]  = (j < kM) ? e[j] * m2 : 0.0f;
                    rhs[j] = (rhs[j] - e[j-1] * rhs[j-1]) * m2;
                }
                dx[kN-1] = rhs[kN-1];
                #pragma unroll
                for (int j = kN - 2; j >= 0; --j)
                    dx[j] = rhs[j] - cp[j] * dx[j+1];
            }

            // ds, dz and fraction-to-boundary alpha.
            // min(-v/dv) tracked as a (num,den) pair via cross-multiplication:
            // whole 46-candidate reduction costs ONE v_rcp_f32 at the end.
            float ds[kM], dz[kM];
            float bn = 1.0f, bd = 0.0f;               // (1,0) == +inf
            #pragma unroll
            for (int k = 0; k < kM; ++k) {
                ds[k] = -rp[k] - (dx[k+1] - dx[k]);
                dz[k] = (-rc[k] - z[k] * ds[k]) * rs[k];
                if (ds[k] < 0.0f && s[k] * bd < bn * (-ds[k])) { bn = s[k]; bd = -ds[k]; }
                if (dz[k] < 0.0f && z[k] * bd < bn * (-dz[k])) { bn = z[k]; bd = -dz[k]; }
            }
            // bd==0 -> rcp(0)=+inf -> alpha=1 (branchless; bn>0 always)
            float alpha = fminf(1.0f, 0.99f * bn * rcp(bd));

            #pragma unroll
            for (int j = 0; j < kN; ++j) x[j] += alpha * dx[j];
            #pragma unroll
            for (int k = 0; k < kM; ++k) { s[k] += alpha * ds[k]; z[k] += alpha * dz[k]; }
        }
    }

    {   // 6 x float4 per lane -> global_store_b128
        float4* ov = (float4*)(out + base);
        #pragma unroll
        for (int v = 0; v < 6; ++v) {
            float4 a;
            a.x = x[4*v+0]; a.y = x[4*v+1]; a.z = x[4*v+2]; a.w = x[4*v+3];
            ov[v] = a;
        }
    }
}

extern "C" void kernel_launch(void* const* d_in, const int* in_sizes, int n_in,
                              void* d_out, int out_size, void* d_ws, size_t ws_size,
                              hipStream_t stream) {
    (void)n_in; (void)d_ws; (void)ws_size; (void)out_size;
    const float* x  = (const float*)d_in[0];   // "x"
    const float* mu = (const float*)d_in[1];   // "mu"
    const float* sg = (const float*)d_in[2];   // "sigma"
    float* out = (float*)d_out;

    const int B = in_sizes[0] / kN;            // 4096
    const int grid = (B + kBlk - 1) / kBlk;
    stoch_opt_ipm_kernel<<<grid, kBlk, 0, stream>>>(x, mu, sg, out, B);
}